// Model_and_SSIM_Guided_48567490183605
// MI455X (gfx1250) — compile-verified
//
#include <hip/hip_runtime.h>
#include <hip/hip_bf16.h>
#include <math.h>
#include <stdint.h>

// ---------------------------------------------------------------------------
// MI455X (gfx1250) implementation of the Mamba + SSIM/SAM-guided fusion model.
// f32 end-to-end (model is latency-bound: ~3 GFLOP / ~120MB, fits in 192MB L2),
// all channel-mixing GEMMs on V_WMMA_F32_16X16X4_F32 with the B tile staged
// into LDS by the Tensor Data Mover (tensor_load_to_lds + s_wait_tensorcnt),
// SSM recurrence via LDS parallel scan, top-k via 128KB-LDS bitonic sort.
// ---------------------------------------------------------------------------

typedef __attribute__((ext_vector_type(2))) float v2f;
typedef __attribute__((ext_vector_type(8))) float v8f;
typedef __attribute__((ext_vector_type(4))) unsigned int u32x4;
typedef __attribute__((ext_vector_type(4))) int i32x4t;
typedef __attribute__((ext_vector_type(8))) int i32x8t;

#define ACT_NONE     0
#define ACT_SILU     1
#define ACT_SOFTPLUS 2
#define ACT_GELU     3
#define ACT_LEAKY    4

static inline int ceildiv(int a, int b) { return (a + b - 1) / b; }

__device__ __forceinline__ float actf(float x, int act) {
    switch (act) {
        case ACT_SILU:     return x / (1.f + __expf(-x));
        case ACT_SOFTPLUS: return (x > 20.f) ? x : log1pf(__expf(x));
        case ACT_GELU:     return 0.5f * x * (1.f + erff(x * 0.70710678118654752f));
        case ACT_LEAKY:    return x > 0.f ? x : 0.1f * x;
        default:           return x;
    }
}

// ------------------------- WMMA f32 GEMM ------------------------------------
// C[M,N] = act(A[M,K] @ W[N,K]^T + bias) (+ addsrc). One wave per 16x16 tile.
// B tile (16 x K, row stride ldw) is DMA'd into LDS by the TDM; A fragments
// are branchless clamped global loads so EXEC stays all-ones for WMMA.
__global__ void k_gemm(const float* __restrict__ A, int lda,
                       const float* __restrict__ W, int ldw,
                       const float* __restrict__ bias,
                       const float* __restrict__ addsrc, int ldadd,
                       float* __restrict__ C, int ldc,
                       int Mmax, const int* __restrict__ Mptr,
                       int N, int K, int act)
{
    int M = Mptr ? *Mptr : Mmax;
    if (M > Mmax) M = Mmax;
    int row0 = blockIdx.x * 16;
    int col0 = blockIdx.y * 16;
    if (row0 >= M) return;                 // uniform across wave (EXEC stays full)
    int lane  = threadIdx.x;               // 0..31, wave32
    int rlo   = lane & 15;
    int khalf = (lane >> 4) * 2;           // 0 or 2

#if defined(__gfx1250__) && __has_builtin(__builtin_amdgcn_wmma_f32_16x16x4_f32)
    __shared__ float sW[16 * 128];         // staged B tile (K <= 128)

#if __has_builtin(__builtin_amdgcn_tensor_load_to_lds) && __has_builtin(__builtin_amdgcn_s_wait_tensorcnt)
    {
        // Tensor DMA descriptor (D#): 2D tile, 4-byte elements,
        // tile = (K x 16 rows), tensor_dim1 = N-col0 so OOB rows read as zero.
        unsigned long long gaddr =
            (unsigned long long)(uintptr_t)(const void*)(W + (size_t)col0 * ldw);
        unsigned lds_off = (unsigned)(uintptr_t)(void*)sW;   // low 32 bits = LDS byte offset
        unsigned rows = (unsigned)(N - col0); if (rows > 16u) rows = 16u;
        unsigned uK = (unsigned)K, uS = (unsigned)ldw;
        u32x4 g0;
        g0[0] = 1u;                                           // count=1, normal mode
        g0[1] = lds_off;                                      // lds_addr (bytes)
        g0[2] = (unsigned)(gaddr & 0xffffffffu);              // global_addr[31:0]
        g0[3] = (unsigned)((gaddr >> 32) & 0x01ffffffu) | (2u << 30);  // addr[56:32] | type=2
        i32x8t g1;
        g1[0] = (int)(2u << 16);                              // data_size=2 (4 bytes)
        g1[1] = (int)(uK << 16);                              // tensor_dim0[15:0] @ bits 63:48
        g1[2] = (int)((uK >> 16) | (rows << 16));             // dim0 hi | tensor_dim1[15:0]
        g1[3] = (int)((rows >> 16) | (uK << 16));             // dim1 hi | tile_dim0 = K
        g1[4] = 16;                                           // tile_dim1 = 16, tile_dim2 = 0
        g1[5] = (int)uS;                                      // tensor_dim0_stride[31:0]
        g1[6] = 0;                                            // stride hi | dim1_stride lo
        g1[7] = 0;
        i32x4t gz4 = {0, 0, 0, 0};
        i32x8t gz8 = {0, 0, 0, 0, 0, 0, 0, 0};
        __builtin_amdgcn_tensor_load_to_lds(g0, g1, gz4, gz4, gz8, 0);
        __builtin_amdgcn_s_wait_tensorcnt((short)0);
    }
#else
    for (int i = lane; i < 16 * K; i += 32) {
        int r = i / K, c = i % K;
        sW[i] = (col0 + r < N) ? W[(size_t)(col0 + r) * ldw + c] : 0.f;
    }
    __syncthreads();
#endif

    v8f acc = {};
    int ar  = row0 + rlo;
    int arc = (ar < M) ? ar : (M - 1);
    bool arow_ok = (ar < M);
    for (int kk = 0; kk < K; kk += 4) {
        int k0 = kk + khalf;
        int k1 = k0 + 1;
        int k0c = (k0 < K) ? k0 : (K - 1);
        int k1c = (k1 < K) ? k1 : (K - 1);
        // branchless clamped loads (no EXEC manipulation around WMMA)
        float a0 = A[(size_t)arc * lda + k0c];
        float a1 = A[(size_t)arc * lda + k1c];
        float b0 = sW[rlo * K + k0c];
        float b1 = sW[rlo * K + k1c];
        v2f a, b;
        a.x = (arow_ok && k0 < K) ? a0 : 0.f;
        a.y = (arow_ok && k1 < K) ? a1 : 0.f;
        b.x = (k0 < K) ? b0 : 0.f;
        b.y = (k1 < K) ? b1 : 0.f;
        acc = __builtin_amdgcn_wmma_f32_16x16x4_f32(false, a, false, b,
                                                    (short)0, acc, false, false);
    }
    int rbase = row0 + (lane >> 4) * 8;
    int c = col0 + rlo;
    if (c < N) {
        float bv = bias ? bias[c] : 0.f;
#pragma unroll
        for (int r = 0; r < 8; ++r) {
            int rr = rbase + r;
            if (rr < M) {
                float v = actf(acc[r] + bv, act);
                if (addsrc) v += addsrc[(size_t)rr * ldadd + c];
                C[(size_t)rr * ldc + c] = v;
            }
        }
    }
#else
    // scalar fallback (host pass / missing builtin)
    int rbase = row0 + (lane >> 4) * 8;
    int c = col0 + rlo;
    if (c < N) {
        float bv = bias ? bias[c] : 0.f;
        for (int r = 0; r < 8; ++r) {
            int rr = rbase + r;
            if (rr < M) {
                float s = 0.f;
                for (int kk = 0; kk < K; ++kk)
                    s += A[(size_t)rr * lda + kk] * W[(size_t)c * ldw + kk];
                float v = actf(s + bv, act);
                if (addsrc) v += addsrc[(size_t)rr * ldadd + c];
                C[(size_t)rr * ldc + c] = v;
            }
        }
    }
#endif
}

// ------------------------- LayerNorm (per token) ----------------------------
__global__ void k_ln(const float* __restrict__ in, int ldin,
                     const float* __restrict__ g, const float* __restrict__ b,
                     float* __restrict__ out, int ldout, int M, int C)
{
    int i = blockIdx.x * blockDim.x + threadIdx.x;
    if (i >= M) return;
    const float* row = in + (size_t)i * ldin;
    float m = 0.f;
    for (int c = 0; c < C; ++c) m += row[c];
    m /= C;
    float v = 0.f;
    for (int c = 0; c < C; ++c) { float d = row[c] - m; v += d * d; }
    v /= C;
    float inv = rsqrtf(v + 1e-5f);
    float* orow = out + (size_t)i * ldout;
    for (int c = 0; c < C; ++c) orow[c] = (row[c] - m) * inv * g[c] + b[c];
}

// ------------------------- depthwise conv (token layout) --------------------
__global__ void k_dwconv(const float* __restrict__ in, int ldin,
                         const float* __restrict__ w, const float* __restrict__ bias,
                         float* __restrict__ out, int ldout,
                         int Hi, int Wi, int C, int KH, int KW,
                         int stride, int pad, int act)
{
    int Ho = (Hi + 2 * pad - KH) / stride + 1;
    int Wo = (Wi + 2 * pad - KW) / stride + 1;
    int idx = blockIdx.x * blockDim.x + threadIdx.x;
    if (idx >= Ho * Wo * C) return;
    int c = idx % C, p = idx / C;
    int wo = p % Wo, ho = p / Wo;
    float s = bias ? bias[c] : 0.f;
    for (int kh = 0; kh < KH; ++kh) {
        int hi = ho * stride - pad + kh;
        if (hi < 0 || hi >= Hi) continue;
        for (int kw = 0; kw < KW; ++kw) {
            int wi = wo * stride - pad + kw;
            if (wi < 0 || wi >= Wi) continue;
            s += in[((size_t)hi * Wi + wi) * ldin + c] * w[(size_t)c * KH * KW + kh * KW + kw];
        }
    }
    out[(size_t)p * ldout + c] = actf(s, act);
}

// ------------------------- dense conv (token layout) ------------------------
__global__ void k_conv(const float* __restrict__ in, const float* __restrict__ w,
                       const float* __restrict__ bias, float* __restrict__ out,
                       int Hi, int Wi, int Cin, int Cout, int KH, int KW,
                       int stride, int pad, int act)
{
    int Ho = (Hi + 2 * pad - KH) / stride + 1;
    int Wo = (Wi + 2 * pad - KW) / stride + 1;
    int idx = blockIdx.x * blockDim.x + threadIdx.x;
    if (idx >= Ho * Wo * Cout) return;
    int co = idx % Cout, p = idx / Cout;
    int wo = p % Wo, ho = p / Wo;
    float s = bias ? bias[co] : 0.f;
    for (int kh = 0; kh < KH; ++kh) {
        int hi = ho * stride - pad + kh;
        if (hi < 0 || hi >= Hi) continue;
        for (int kw = 0; kw < KW; ++kw) {
            int wi = wo * stride - pad + kw;
            if (wi < 0 || wi >= Wi) continue;
            const float* ip = in + ((size_t)hi * Wi + wi) * Cin;
            const float* wp = w + (((size_t)co * Cin) * KH + kh) * KW + kw;
            for (int ci = 0; ci < Cin; ++ci) s += ip[ci] * wp[(size_t)ci * KH * KW];
        }
    }
    out[(size_t)p * Cout + co] = actf(s, act);
}

// ------------------------- build the 4 scan directions ----------------------
__global__ void k_make_xs(const float* __restrict__ xi, float* __restrict__ xs,
                          int H, int W, int C)
{
    int L = H * W;
    int idx = blockIdx.x * blockDim.x + threadIdx.x;
    if (idx >= 4 * L * C) return;
    int c = idx % C;
    int t = (idx / C) % L;
    int k = idx / (C * L);
    int src;
    if (k == 0) src = t;
    else if (k == 1) { int w = t / H, h = t % H; src = h * W + w; }
    else if (k == 2) src = L - 1 - t;
    else { int t2 = L - 1 - t; int w = t2 / H, h = t2 % H; src = h * W + w; }
    xs[idx] = xi[(size_t)src * C + c];
}

// ------------------------- SSM parallel scan --------------------------------
#define SCAN_T 256
__global__ void k_scan(const float* __restrict__ xs,     // [4][L][C]
                       const float* __restrict__ delta,  // [4][L][C]
                       const float* __restrict__ dbl,    // [4][L][dbl_ld]
                       int dbl_ld, int bo, int co,
                       const float* __restrict__ A_log,  // [4][C][n]
                       const float* __restrict__ Dp,     // [4][C]
                       float* __restrict__ ys,           // [4][L][C]
                       int L, int C, int n, int chunk)
{
    int d = blockIdx.x, k = blockIdx.y;
    int t = threadIdx.x;
    const float* xk   = xs    + (size_t)k * L * C;
    const float* dk   = delta + (size_t)k * L * C;
    const float* dblk = dbl   + (size_t)k * L * dbl_ld;
    __shared__ float sA[SCAN_T], sB[SCAN_T];
    float yacc[64];
    for (int j = 0; j < chunk; ++j) yacc[j] = 0.f;
    float Dval = Dp[k * C + d];
    int l0 = t * chunk;
    for (int in_ = 0; in_ < n; ++in_) {
        float Acoef = -__expf(A_log[((size_t)k * C + d) * n + in_]);
        float Ap = 1.f, Bs = 0.f;
        for (int j = 0; j < chunk; ++j) {
            int l = l0 + j;
            if (l < L) {
                __builtin_prefetch(&dk[(size_t)(l + 8) * C + d], 0, 1);
                float dl = dk[(size_t)l * C + d];
                float u  = xk[(size_t)l * C + d];
                float a  = __expf(dl * Acoef);
                float bv = dl * dblk[(size_t)l * dbl_ld + bo + in_] * u;
                Ap = Ap * a;
                Bs = Bs * a + bv;
            }
        }
        sA[t] = Ap; sB[t] = Bs;
        __syncthreads();
        for (int off = 1; off < SCAN_T; off <<= 1) {
            float a2 = sA[t], b2 = sB[t];
            float a1 = 1.f, b1 = 0.f;
            if (t >= off) { a1 = sA[t - off]; b1 = sB[t - off]; }
            __syncthreads();
            sA[t] = a1 * a2;
            sB[t] = b1 * a2 + b2;
            __syncthreads();
        }
        float hprev = (t > 0) ? sB[t - 1] : 0.f;
        float h = hprev;
        for (int j = 0; j < chunk; ++j) {
            int l = l0 + j;
            if (l < L) {
                float dl = dk[(size_t)l * C + d];
                float u  = xk[(size_t)l * C + d];
                float a  = __expf(dl * Acoef);
                float bv = dl * dblk[(size_t)l * dbl_ld + bo + in_] * u;
                h = a * h + bv;
                yacc[j] += h * dblk[(size_t)l * dbl_ld + co + in_];
            }
        }
        __syncthreads();
    }
    for (int j = 0; j < chunk; ++j) {
        int l = l0 + j;
        if (l < L)
            ys[((size_t)k * L + l) * C + d] = yacc[j] + xk[(size_t)l * C + d] * Dval;
    }
}

// ------------------------- combine 4 directional outputs --------------------
__global__ void k_combine(const float* __restrict__ ys, float* __restrict__ yc,
                          int H, int W, int C)
{
    int L = H * W;
    int idx = blockIdx.x * blockDim.x + threadIdx.x;
    if (idx >= L * C) return;
    int c = idx % C, l = idx / C;
    int h = l / W, w = l % W;
    int m = w * H + h;
    yc[idx] = ys[((size_t)0 * L + l) * C + c]
            + ys[((size_t)1 * L + m) * C + c]
            + ys[((size_t)2 * L + (L - 1 - l)) * C + c]
            + ys[((size_t)3 * L + (L - 1 - m)) * C + c];
}

__global__ void k_mulsilu(float* __restrict__ y, int ldy,
                          const float* __restrict__ z, int ldz, int M, int C)
{
    int idx = blockIdx.x * blockDim.x + threadIdx.x;
    if (idx >= M * C) return;
    int c = idx % C, i = idx / C;
    float zz = z[(size_t)i * ldz + c];
    y[(size_t)i * ldy + c] *= zz / (1.f + __expf(-zz));
}

// ------------------------- layout transforms --------------------------------
__global__ void k_chw2tok(const float* __restrict__ src, float* __restrict__ dst,
                          int C, int H, int W)
{
    int HW = H * W;
    int t = blockIdx.x * blockDim.x + threadIdx.x;
    if (t >= C * HW) return;
    int c = t % C, l = t / C;
    dst[t] = src[(size_t)c * HW + l];
}
__global__ void k_tok2chw(const float* __restrict__ src, float* __restrict__ dst,
                          int C, int H, int W)
{
    int HW = H * W;
    int t = blockIdx.x * blockDim.x + threadIdx.x;
    if (t >= C * HW) return;
    int c = t / HW, l = t % HW;
    dst[t] = src[(size_t)l * C + c];
}
__global__ void k_concat2(float* __restrict__ dst, const float* __restrict__ a, int Ca,
                          const float* __restrict__ b, int Cb, int M)
{
    int C = Ca + Cb;
    int t = blockIdx.x * blockDim.x + threadIdx.x;
    if (t >= M * C) return;
    int c = t % C, i = t / C;
    dst[t] = (c < Ca) ? a[(size_t)i * Ca + c] : b[(size_t)i * Cb + (c - Ca)];
}
__global__ void k_pixshuf(const float* __restrict__ in, float* __restrict__ out,
                          int Hs, int Ws, int C, int r)
{
    int Wo = Ws * r, Ho = Hs * r;
    int t = blockIdx.x * blockDim.x + threadIdx.x;
    if (t >= Ho * Wo * C) return;
    int co = t % C, p = t / C;
    int wo = p % Wo, ho = p / Wo;
    int h = ho / r, rh = ho % r, w = wo / r, rw = wo % r;
    out[t] = in[(size_t)(h * Ws + w) * (C * r * r) + (size_t)co * r * r + rh * r + rw];
}
__global__ void k_add(const float* __restrict__ a, const float* __restrict__ b,
                      float* __restrict__ o, int n)
{
    int t = blockIdx.x * blockDim.x + threadIdx.x;
    if (t < n) o[t] = a[t] + b[t];
}

// ------------------------- PMSI / SSIM / SAM --------------------------------
__global__ void k_pmsi(const float* __restrict__ pre, const float* __restrict__ Rm,
                       float* __restrict__ pm, int L)
{
    int t = blockIdx.x * blockDim.x + threadIdx.x;
    if (t >= 3 * L) return;
    int r = t / L, l = t % L;
    float s = 0.f;
    for (int c = 0; c < 31; ++c) s += pre[(size_t)l * 31 + c] * Rm[r * 31 + c];
    pm[t] = fminf(1.f, fmaxf(0.f, s));
}

__global__ void k_ssim(const float* __restrict__ x, const float* __restrict__ y,
                       float* __restrict__ out, int H, int W)
{
    int l = blockIdx.x * blockDim.x + threadIdx.x;
    if (l >= H * W) return;
    int h = l / W, w0 = l % W;
    float g[5]; float gs = 0.f;
    for (int i = 0; i < 5; ++i) { g[i] = __expf(-(float)((i - 2) * (i - 2)) / 4.5f); gs += g[i]; }
    for (int i = 0; i < 5; ++i) g[i] /= gs;
    const float C1 = 1e-4f, C2 = 9e-4f;
    float msum = 0.f;
    for (int c = 0; c < 3; ++c) {
        const float* xc = x + (size_t)c * H * W;
        const float* yc = y + (size_t)c * H * W;
        float mu1 = 0, mu2 = 0, sx = 0, sy = 0, sxy = 0;
        for (int i = 0; i < 5; ++i) {
            int hh = h + i - 2;
            if (hh < 0 || hh >= H) continue;
            for (int j = 0; j < 5; ++j) {
                int ww = w0 + j - 2;
                if (ww < 0 || ww >= W) continue;
                float wgt = g[i] * g[j];
                float xv = xc[(size_t)hh * W + ww], yv = yc[(size_t)hh * W + ww];
                mu1 += wgt * xv; mu2 += wgt * yv;
                sx += wgt * xv * xv; sy += wgt * yv * yv; sxy += wgt * xv * yv;
            }
        }
        float s1 = sx - mu1 * mu1, s2 = sy - mu2 * mu2, s12 = sxy - mu1 * mu2;
        msum += ((2.f * mu1 * mu2 + C1) * (2.f * s12 + C2)) /
                ((mu1 * mu1 + mu2 * mu2 + C1) * (s1 + s2 + C2));
    }
    out[l] = msum * (1.f / 3.f);
}

__global__ void k_samdeg(const float* __restrict__ plr, const float* __restrict__ gt,
                         float* __restrict__ out, int Ls, int C)
{
    int l = blockIdx.x * blockDim.x + threadIdx.x;
    if (l >= Ls) return;
    float dot = 0, na = 0, nb = 0;
    for (int c = 0; c < C; ++c) {
        float a = plr[(size_t)l * C + c];
        float b = gt[(size_t)c * Ls + l];
        dot += a * b; na += a * a; nb += b * b;
    }
    const float eps = 2.2204e-16f;
    float v = (dot + eps) / (sqrtf(na) * sqrtf(nb) + eps);
    v = fminf(1.f, fmaxf(-1.f, v));
    out[l] = acosf(v) * 57.295779513082320877f;
}

// ------------------------- reduction + k1/k2 --------------------------------
__global__ void k_reduce(const float* __restrict__ x, int n, float* __restrict__ part)
{
    __shared__ float s[256];
    float acc = 0.f;
    for (int i = blockIdx.x * 256 + threadIdx.x; i < n; i += gridDim.x * 256) acc += x[i];
    s[threadIdx.x] = acc;
    __syncthreads();
    for (int o = 128; o > 0; o >>= 1) {
        if (threadIdx.x < o) s[threadIdx.x] += s[threadIdx.x + o];
        __syncthreads();
    }
    if (threadIdx.x == 0) part[blockIdx.x] = s[0];
}
__global__ void k_finalize(const float* __restrict__ part, int nparts, int n,
                           int* __restrict__ kv, int N1, int N2, int K1cap, int K2cap)
{
    if (threadIdx.x != 0 || blockIdx.x != 0) return;
    float s = 0.f;
    for (int i = 0; i < nparts; ++i) s += part[i];
    float ratio = 0.3f * (1.f - s / (float)n);
    if (ratio < 0.f) ratio = 0.f;
    int k1 = (int)((float)N1 * ratio); if (k1 > K1cap) k1 = K1cap; if (k1 < 0) k1 = 0;
    int k2 = (int)((float)N2 * ratio); if (k2 > K2cap) k2 = K2cap; if (k2 < 0) k2 = 0;
    kv[0] = k1; kv[1] = k2;
}

// ------------------------- bitonic argsort in LDS ---------------------------
__global__ void k_sort(const float* __restrict__ score, int N, int desc,
                       int* __restrict__ idx_out, int KOUT)
{
    extern __shared__ unsigned long long sdat[];
    int t = threadIdx.x, T = blockDim.x;
    for (int i = t; i < N; i += T) {
        unsigned u = __float_as_uint(score[i]);
        u = (u & 0x80000000u) ? ~u : (u | 0x80000000u);  // order-preserving map
        if (desc) u = ~u;
        sdat[i] = ((unsigned long long)u << 32) | (unsigned)i;
    }
    __syncthreads();
    for (int ks = 2; ks <= N; ks <<= 1) {
        for (int j = ks >> 1; j > 0; j >>= 1) {
            for (int i = t; i < N; i += T) {
                int ixj = i ^ j;
                if (ixj > i) {
                    bool up = ((i & ks) == 0);
                    unsigned long long a = sdat[i], b = sdat[ixj];
                    bool sw = up ? (a > b) : (a < b);
                    if (sw) { sdat[i] = b; sdat[ixj] = a; }
                }
            }
            __syncthreads();
        }
    }
    for (int i = t; i < KOUT && i < N; i += T) idx_out[i] = (int)(sdat[i] & 0xffffffffu);
}

// ------------------------- gather/scatter -----------------------------------
__global__ void k_gather(const float* __restrict__ tok, int ld, const int* __restrict__ idx,
                         const int* __restrict__ kptr, int KMAX, int C, float* __restrict__ out)
{
    int t = blockIdx.x * blockDim.x + threadIdx.x;
    if (t >= KMAX * C) return;
    int c = t % C, i = t / C;
    int k = *kptr;
    out[t] = (i < k) ? tok[(size_t)idx[i] * ld + c] : 0.f;
}
__global__ void k_scatter_add(float* __restrict__ dst, int ld, const int* __restrict__ idx,
                              const int* __restrict__ kptr, int KMAX, int C,
                              const float* __restrict__ v)
{
    int t = blockIdx.x * blockDim.x + threadIdx.x;
    if (t >= KMAX * C) return;
    int c = t % C, i = t / C;
    if (i < *kptr) dst[(size_t)idx[i] * ld + c] += v[(size_t)i * C + c];
}

// ------------------------- transformer pieces -------------------------------
__global__ void k_l2norms(const float* __restrict__ q, const float* __restrict__ kk,
                          int ld, const int* __restrict__ kptr, int KMAX, int F,
                          float* __restrict__ norms)
{
    int t = threadIdx.x;          // block of 64
    int kact = *kptr;
    if (kact > KMAX) kact = KMAX;
    if (t < F) {
        float s = 0.f;
        for (int i = 0; i < kact; ++i) { float v = q[(size_t)i * ld + t]; s += v * v; }
        norms[t] = sqrtf(s);
    } else if (t >= 32 && t - 32 < F) {
        int c = t - 32;
        float s = 0.f;
        for (int i = 0; i < kact; ++i) { float v = kk[(size_t)i * ld + c]; s += v * v; }
        norms[F + c] = sqrtf(s);
    }
}
__global__ void k_attn(const float* __restrict__ Q, const float* __restrict__ K,
                       int ld, const int* __restrict__ kptr, int KMAX,
                       const float* __restrict__ norms, const float* __restrict__ rescale,
                       float* __restrict__ attn)
{
    __shared__ float slog[196];
    int t = threadIdx.x;          // block of 256
    int kact = *kptr;
    if (kact > KMAX) kact = KMAX;
    if (t < 196) {
        int h = t / 49, rem = t % 49, d = rem / 7, e = rem % 7;
        float s = 0.f;
        for (int i = 0; i < kact; ++i)
            s += K[(size_t)i * ld + h * 7 + d] * Q[(size_t)i * ld + h * 7 + e];
        float qn = fmaxf(norms[h * 7 + e], 1e-12f);
        float kn = fmaxf(norms[28 + h * 7 + d], 1e-12f);
        slog[t] = s / (qn * kn) * rescale[h];
    }
    __syncthreads();
    if (t < 28) {
        int h = t / 7, d = t % 7, base = h * 49 + d * 7;
        float mx = -1e30f;
        for (int e = 0; e < 7; ++e) mx = fmaxf(mx, slog[base + e]);
        float ex[7], se = 0.f;
        for (int e = 0; e < 7; ++e) { ex[e] = __expf(slog[base + e] - mx); se += ex[e]; }
        for (int e = 0; e < 7; ++e) attn[base + e] = ex[e] / se;
    }
}
__global__ void k_attn_apply(const float* __restrict__ attn, const float* __restrict__ V,
                             int ld, float* __restrict__ xo, int KMAX)
{
    int t = blockIdx.x * blockDim.x + threadIdx.x;
    if (t >= KMAX * 28) return;
    int i = t / 28, j = t % 28;
    int h = j / 7, d = j % 7;
    float s = 0.f;
    for (int e = 0; e < 7; ++e)
        s += attn[h * 49 + d * 7 + e] * V[(size_t)i * ld + h * 7 + e];
    xo[t] = s;
}

// ===========================================================================
// Host orchestration
// ===========================================================================

constexpr int    K1MAX = 8192;
constexpr int    K2MAX = 160;
constexpr size_t LTOK  = 16384;

constexpr size_t o_fufea   = 0;
constexpr size_t o_gtmsi   = o_fufea + LTOK * 31;
constexpr size_t o_gthsi   = o_gtmsi + LTOK * 3;
constexpr size_t o_m64     = o_gthsi + 256 * 31;
constexpr size_t o_f8a     = o_m64 + LTOK * 31;
constexpr size_t o_f8b     = o_f8a + 32 * 32 * 31;
constexpr size_t o_m8      = o_f8b + 256 * 31;
constexpr size_t o_upbig   = o_m8 + 256 * 31;
constexpr size_t o_up8     = o_upbig + 256 * 1984;
constexpr size_t o_cat     = o_up8 + LTOK * 31;
constexpr size_t o_pre     = o_cat + LTOK * 62;
constexpr size_t o_pmsi    = o_pre + LTOK * 31;
constexpr size_t o_sscore  = o_pmsi + 3 * LTOK;
constexpr size_t o_part    = o_sscore + LTOK;
constexpr size_t o_fap     = o_part + 256;
constexpr size_t o_ssimout = o_fap + LTOK * 31;
constexpr size_t o_plr     = o_ssimout + LTOK * 31;
constexpr size_t o_samsc   = o_plr + 256 * 31;
constexpr size_t o_samfea  = o_samsc + 256;
constexpr size_t o_samouts = o_samfea + 256 * 31;
constexpr size_t o_samout  = o_samouts + 256 * 31;
constexpr size_t o_fuall   = o_samout + LTOK * 31;
constexpr size_t o_m2      = o_fuall + LTOK * 31;
constexpr size_t o_re      = o_m2 + LTOK * 31;
constexpr size_t o_outtok  = o_re + LTOK * 31;
constexpr size_t o_v1      = o_outtok + LTOK * 31;
constexpr size_t o_v1r     = o_v1 + (size_t)K1MAX * 31;
constexpr size_t o_v2      = o_v1r + (size_t)K1MAX * 31;
constexpr size_t o_v2r     = o_v2 + (size_t)K2MAX * 31;
constexpr size_t o_tws     = o_v2r + (size_t)K2MAX * 31;
constexpr size_t o_ints    = o_tws + (size_t)K1MAX * 283 + 320;
constexpr size_t o_arena   = o_ints + (size_t)(K1MAX + K2MAX + 8);
// arena needs LTOK * 1151 floats; total ws ≈ 120 MB.

static void gemm(hipStream_t s, const float* A, int lda, const float* W, int ldw,
                 const float* bias, const float* add, int ldadd,
                 float* out, int ldc, int Mmax, const int* Mptr, int N, int K, int act)
{
    dim3 g(ceildiv(Mmax, 16), ceildiv(N, 16));
    k_gemm<<<g, 32, 0, s>>>(A, lda, W, ldw, bias, add, ldadd, out, ldc, Mmax, Mptr, N, K, act);
}

struct MambaP {
    const float *ln_g, *ln_b, *in_w, *conv_w, *conv_b, *x_w, *dt_w, *dt_b,
                *A_log, *D, *on_g, *on_b, *out_w;
};
static MambaP mk_mamba(void* const* din, int b) {
    MambaP p;
    p.ln_g = (const float*)din[b + 0];  p.ln_b  = (const float*)din[b + 1];
    p.in_w = (const float*)din[b + 2];  p.conv_w = (const float*)din[b + 3];
    p.conv_b = (const float*)din[b + 4]; p.x_w   = (const float*)din[b + 5];
    p.dt_w = (const float*)din[b + 6];  p.dt_b  = (const float*)din[b + 7];
    p.A_log = (const float*)din[b + 8]; p.D     = (const float*)din[b + 9];
    p.on_g = (const float*)din[b + 10]; p.on_b  = (const float*)din[b + 11];
    p.out_w = (const float*)din[b + 12];
    return p;
}

static void run_mamba(hipStream_t s, const float* xin, int H, int W,
                      const MambaP& p, float* arena, float* out_tok)
{
    int L = H * W;
    float* xn  = arena;
    float* xz  = xn  + (size_t)L * 31;
    float* xi  = xz  + (size_t)L * 124;
    float* xs  = xi  + (size_t)L * 62;
    float* dbl = xs  + (size_t)4 * L * 62;
    float* dl  = dbl + (size_t)4 * L * 32;
    float* ysb = dl  + (size_t)4 * L * 62;
    float* yc  = ysb + (size_t)4 * L * 62;

    k_ln<<<ceildiv(L, 256), 256, 0, s>>>(xin, 31, p.ln_g, p.ln_b, xn, 31, L, 31);
    gemm(s, xn, 31, p.in_w, 31, nullptr, nullptr, 0, xz, 124, L, nullptr, 124, 31, ACT_NONE);
    k_dwconv<<<ceildiv(L * 62, 256), 256, 0, s>>>(xz, 124, p.conv_w, p.conv_b,
                                                  xi, 62, H, W, 62, 3, 3, 1, 1, ACT_SILU);
    k_make_xs<<<ceildiv(4 * L * 62, 256), 256, 0, s>>>(xi, xs, H, W, 62);
    for (int k = 0; k < 4; ++k)
        gemm(s, xs + (size_t)k * L * 62, 62, p.x_w + (size_t)k * 32 * 62, 62,
             nullptr, nullptr, 0, dbl + (size_t)k * L * 32, 32, L, nullptr, 32, 62, ACT_NONE);
    for (int k = 0; k < 4; ++k)
        gemm(s, dbl + (size_t)k * L * 32, 32, p.dt_w + (size_t)k * 62 * 2, 2,
             p.dt_b + (size_t)k * 62, nullptr, 0, dl + (size_t)k * L * 62, 62,
             L, nullptr, 62, 2, ACT_SOFTPLUS);
    int chunk = ceildiv(L, SCAN_T);
    k_scan<<<dim3(62, 4), SCAN_T, 0, s>>>(xs, dl, dbl, 32, 2, 17, p.A_log, p.D,
                                          ysb, L, 62, 15, chunk);
    k_combine<<<ceildiv(L * 62, 256), 256, 0, s>>>(ysb, yc, H, W, 62);
    k_ln<<<ceildiv(L, 256), 256, 0, s>>>(yc, 62, p.on_g, p.on_b, yc, 62, L, 62);
    k_mulsilu<<<ceildiv(L * 62, 256), 256, 0, s>>>(yc, 62, xz + 62, 124, L, 62);
    gemm(s, yc, 62, p.out_w, 62, nullptr, xin, 31, out_tok, 31, L, nullptr, 31, 62, ACT_NONE);
}

struct TransP { const float* p[16]; };
enum { T_LN1G, T_LN1B, T_LN2G, T_LN2B, T_WQ, T_WK, T_WV, T_RESC,
       T_PROJW, T_PROJB, T_PE1W, T_PE1B, T_PE2W, T_PE2B, T_FC1W, T_FC2W };
static TransP mk_trans(void* const* din, int b) {
    TransP t;
    for (int i = 0; i < 16; ++i) t.p[i] = (const float*)din[b + i];
    return t;
}

static void run_transformer(hipStream_t s, const TransP& T, const float* vin, int KMAX,
                            const int* kptr, float* vout, float* tws)
{
    float* xln = tws;
    float* q   = xln + (size_t)KMAX * 31;
    float* kk  = q   + (size_t)KMAX * 28;
    float* vv  = kk  + (size_t)KMAX * 28;
    float* xo  = vv  + (size_t)KMAX * 28;
    float* y1  = xo  + (size_t)KMAX * 28;
    float* t31 = y1  + (size_t)KMAX * 31;
    float* y   = t31 + (size_t)KMAX * 31;
    float* yn  = y   + (size_t)KMAX * 31;
    float* h15 = yn  + (size_t)KMAX * 31;
    float* norms = h15 + (size_t)KMAX * 16;
    float* attn  = norms + 64;
    int nb = ceildiv(KMAX, 256);

    k_ln<<<nb, 256, 0, s>>>(vin, 31, T.p[T_LN1G], T.p[T_LN1B], xln, 31, KMAX, 31);
    gemm(s, xln, 31, T.p[T_WQ], 31, nullptr, nullptr, 0, q,  28, KMAX, nullptr, 28, 31, ACT_NONE);
    gemm(s, xln, 31, T.p[T_WK], 31, nullptr, nullptr, 0, kk, 28, KMAX, nullptr, 28, 31, ACT_NONE);
    gemm(s, xln, 31, T.p[T_WV], 31, nullptr, nullptr, 0, vv, 28, KMAX, nullptr, 28, 31, ACT_NONE);
    k_l2norms<<<1, 64, 0, s>>>(q, kk, 28, kptr, KMAX, 28, norms);
    k_attn<<<1, 256, 0, s>>>(q, kk, 28, kptr, KMAX, norms, T.p[T_RESC], attn);
    k_attn_apply<<<ceildiv(KMAX * 28, 256), 256, 0, s>>>(attn, vv, 28, xo, KMAX);
    gemm(s, xo, 28, T.p[T_PROJW], 28, T.p[T_PROJB], vin, 31, y1, 31, KMAX, nullptr, 31, 28, ACT_NONE);
    gemm(s, vv, 28, T.p[T_PE1W], 28, T.p[T_PE1B], nullptr, 0, t31, 31, KMAX, nullptr, 31, 28, ACT_GELU);
    gemm(s, t31, 31, T.p[T_PE2W], 31, T.p[T_PE2B], y1, 31, y, 31, KMAX, nullptr, 31, 31, ACT_NONE);
    k_ln<<<nb, 256, 0, s>>>(y, 31, T.p[T_LN2G], T.p[T_LN2B], yn, 31, KMAX, 31);
    gemm(s, yn, 31, T.p[T_FC1W], 31, nullptr, nullptr, 0, h15, 16, KMAX, nullptr, 15, 31, ACT_LEAKY);
    gemm(s, h15, 16, T.p[T_FC2W], 15, nullptr, y, 31, vout, 31, KMAX, nullptr, 31, 15, ACT_NONE);
}

extern "C" void kernel_launch(void* const* d_in, const int* in_sizes, int n_in,
                              void* d_out, int out_size, void* d_ws, size_t ws_size,
                              hipStream_t stream)
{
    (void)in_sizes; (void)n_in; (void)out_size; (void)ws_size;
    const float* fufea  = (const float*)d_in[0];
    const float* Rm     = (const float*)d_in[1];
    const float* coeff  = (const float*)d_in[2];
    const float* gt_msi = (const float*)d_in[5];
    const float* gt_hsi = (const float*)d_in[6];
    MambaP m64p = mk_mamba(d_in, 7);
    MambaP m8p  = mk_mamba(d_in, 20);
    MambaP mfp  = mk_mamba(d_in, 33);
    MambaP mf2p = mk_mamba(d_in, 46);
    const float* down1_w   = (const float*)d_in[59];
    const float* down2_w   = (const float*)d_in[60];
    const float* up_w      = (const float*)d_in[61];
    const float* up_b      = (const float*)d_in[62];
    const float* mcat_w    = (const float*)d_in[63];
    const float* mcat_b    = (const float*)d_in[64];
    const float* lnorm_g   = (const float*)d_in[65];
    const float* lnorm_b   = (const float*)d_in[66];
    const float* pancat_w  = (const float*)d_in[67];
    const float* pancat_b  = (const float*)d_in[68];
    const float* hsicat_w  = (const float*)d_in[69];
    const float* hsicat_b  = (const float*)d_in[70];
    const float* up2_w     = (const float*)d_in[71];
    const float* up2_b     = (const float*)d_in[72];
    const float* cout_w    = (const float*)d_in[73];
    const float* cout_b    = (const float*)d_in[74];
    const float* cout2_w   = (const float*)d_in[75];
    const float* cout2_b   = (const float*)d_in[76];
    TransP ssimT = mk_trans(d_in, 77);
    TransP samT  = mk_trans(d_in, 93);

    float* w = (float*)d_ws;
    int* idx1 = (int*)(w + o_ints);
    int* idx2 = idx1 + K1MAX;
    int* kv   = idx2 + K2MAX;
    const int L = (int)LTOK;

    // ---- layout transforms for inputs
    k_chw2tok<<<ceildiv(31 * L, 256), 256, 0, stream>>>(fufea,  w + o_fufea, 31, 128, 128);
    k_chw2tok<<<ceildiv(3 * L, 256),  256, 0, stream>>>(gt_msi, w + o_gtmsi, 3, 128, 128);
    k_chw2tok<<<ceildiv(31 * 256, 256), 256, 0, stream>>>(gt_hsi, w + o_gthsi, 31, 16, 16);

    // ---- mamba_64
    run_mamba(stream, w + o_fufea, 128, 128, m64p, w + o_arena, w + o_m64);

    // ---- downsample path + mamba_8 + up
    k_conv<<<ceildiv(32 * 32 * 31, 256), 256, 0, stream>>>(w + o_m64, down1_w, nullptr,
        w + o_f8a, 128, 128, 31, 31, 6, 6, 4, 2, ACT_NONE);
    k_conv<<<ceildiv(256 * 31, 256), 256, 0, stream>>>(w + o_f8a, down2_w, nullptr,
        w + o_f8b, 32, 32, 31, 31, 4, 4, 2, 1, ACT_NONE);
    run_mamba(stream, w + o_f8b, 16, 16, m8p, w + o_arena, w + o_m8);
    gemm(stream, w + o_m8, 31, up_w, 31, up_b, nullptr, 0,
         w + o_upbig, 1984, 256, nullptr, 1984, 31, ACT_NONE);
    k_pixshuf<<<ceildiv(L * 31, 256), 256, 0, stream>>>(w + o_upbig, w + o_up8, 16, 16, 31, 8);

    // ---- fuse + mambafusion + lnorm
    k_concat2<<<ceildiv(L * 62, 256), 256, 0, stream>>>(w + o_cat, w + o_m64, 31, w + o_up8, 31, L);
    gemm(stream, w + o_cat, 62, mcat_w, 62, mcat_b, nullptr, 0,
         w + o_pre, 31, L, nullptr, 31, 62, ACT_NONE);
    run_mamba(stream, w + o_pre, 128, 128, mfp, w + o_arena, w + o_pre);
    k_ln<<<ceildiv(L, 256), 256, 0, stream>>>(w + o_pre, 31, lnorm_g, lnorm_b, w + o_pre, 31, L, 31);

    // ---- PMSI + SSIM score + (k1,k2)
    k_pmsi<<<ceildiv(3 * L, 256), 256, 0, stream>>>(w + o_pre, Rm, w + o_pmsi, L);
    k_ssim<<<ceildiv(L, 256), 256, 0, stream>>>(w + o_pmsi, gt_msi, w + o_sscore, 128, 128);
    k_reduce<<<64, 256, 0, stream>>>(w + o_sscore, L, w + o_part);
    k_finalize<<<1, 1, 0, stream>>>(w + o_part, 64, L, kv, 16384, 256, K1MAX, K2MAX);
    k_sort<<<1, 1024, (size_t)16384 * 8, stream>>>(w + o_sscore, 16384, 0, idx1, K1MAX);

    // ---- fea_add_pan + SSIM refine
    k_concat2<<<ceildiv(L * 34, 256), 256, 0, stream>>>(w + o_cat, w + o_pre, 31, w + o_gtmsi, 3, L);
    gemm(stream, w + o_cat, 34, pancat_w, 34, pancat_b, nullptr, 0,
         w + o_fap, 31, L, nullptr, 31, 34, ACT_NONE);
    k_gather<<<ceildiv(K1MAX * 31, 256), 256, 0, stream>>>(w + o_fap, 31, idx1, kv + 0,
                                                           K1MAX, 31, w + o_v1);
    run_transformer(stream, ssimT, w + o_v1, K1MAX, kv + 0, w + o_v1r, w + o_tws);
    (void)hipMemcpyAsync(w + o_ssimout, w + o_fap, (size_t)L * 31 * sizeof(float),
                         hipMemcpyDeviceToDevice, stream);
    k_scatter_add<<<ceildiv(K1MAX * 31, 256), 256, 0, stream>>>(w + o_ssimout, 31, idx1,
                                                                kv + 0, K1MAX, 31, w + o_v1r);

    // ---- PLR + SAM score + SAM refine
    k_dwconv<<<ceildiv(256 * 31, 256), 256, 0, stream>>>(w + o_ssimout, 31, coeff, nullptr,
        w + o_plr, 31, 128, 128, 31, 9, 9, 8, 4, ACT_NONE);
    k_samdeg<<<1, 256, 0, stream>>>(w + o_plr, gt_hsi, w + o_samsc, 256, 31);
    k_sort<<<1, 128, (size_t)256 * 8, stream>>>(w + o_samsc, 256, 1, idx2, K2MAX);
    k_concat2<<<ceildiv(256 * 62, 256), 256, 0, stream>>>(w + o_cat, w + o_plr, 31,
                                                          w + o_gthsi, 31, 256);
    gemm(stream, w + o_cat, 62, hsicat_w, 62, hsicat_b, nullptr, 0,
         w + o_samfea, 31, 256, nullptr, 31, 62, ACT_NONE);
    k_gather<<<ceildiv(K2MAX * 31, 256), 256, 0, stream>>>(w + o_samfea, 31, idx2, kv + 1,
                                                           K2MAX, 31, w + o_v2);
    run_transformer(stream, samT, w + o_v2, K2MAX, kv + 1, w + o_v2r, w + o_tws);
    (void)hipMemcpyAsync(w + o_samouts, w + o_samfea, (size_t)256 * 31 * sizeof(float),
                         hipMemcpyDeviceToDevice, stream);
    k_scatter_add<<<ceildiv(K2MAX * 31, 256), 256, 0, stream>>>(w + o_samouts, 31, idx2,
                                                                kv + 1, K2MAX, 31, w + o_v2r);
    gemm(stream, w + o_samouts, 31, up2_w, 31, up2_b, nullptr, 0,
         w + o_upbig, 1984, 256, nullptr, 1984, 31, ACT_NONE);
    k_pixshuf<<<ceildiv(L * 31, 256), 256, 0, stream>>>(w + o_upbig, w + o_samout, 16, 16, 31, 8);

    // ---- final fusion
    k_concat2<<<ceildiv(L * 62, 256), 256, 0, stream>>>(w + o_cat, w + o_ssimout, 31,
                                                        w + o_samout, 31, L);
    gemm(stream, w + o_cat, 62, cout_w, 62, cout_b, nullptr, 0,
         w + o_fuall, 31, L, nullptr, 31, 62, ACT_NONE);
    run_mamba(stream, w + o_fuall, 128, 128, mf2p, w + o_arena, w + o_m2);
    k_add<<<ceildiv(L * 31, 256), 256, 0, stream>>>(w + o_m2, w + o_fuall, w + o_re, L * 31);
    k_concat2<<<ceildiv(L * 62, 256), 256, 0, stream>>>(w + o_cat, w + o_re, 31,
                                                        w + o_fufea, 31, L);
    gemm(stream, w + o_cat, 62, cout2_w, 62, cout2_b, nullptr, 0,
         w + o_outtok, 31, L, nullptr, 31, 62, ACT_NONE);
    k_tok2chw<<<ceildiv(31 * L, 256), 256, 0, stream>>>(w + o_outtok, (float*)d_out, 31, 128, 128);
}